// SCF_GRULayer_58420145160916
// MI455X (gfx1250) — compile-verified
//
#include <hip/hip_runtime.h>

// GRU over T=40 steps, BN=8192 rows, H=48, fused into ONE kernel:
//  - block = one batch element b (64 rows = the neighbor group), 4 waves x 16 rows
//  - img mean + img-part of gi (time-invariant) computed once per block
//  - per-step: x-part GEMM (K=20) + h-part GEMM (K=48) via V_WMMA_F32_16X16X4_F32
//  - r/z gates: gi and gh folded into one accumulator chain (init = gi0 + b_hh)
//  - n gate: i_n and h_n kept in separate accumulators (cand = tanh(i_n + r*h_n))

#define T_STEPS 40
#define B_BATCH 128
#define N_GRP   64
#define BN      8192
#define H       48
#define G3      144   // 3*H
#define DIN     52
#define IMG_C   32
#define IMG_HW  6400  // 80*80

typedef __attribute__((ext_vector_type(2))) float v2f;
typedef __attribute__((ext_vector_type(4))) float v4f;
typedef __attribute__((ext_vector_type(8))) float v8f;

static __device__ __forceinline__ v8f wmma4(v2f a, v2f b, v8f c) {
  // D = A(16x4) x B(4x16) + C(16x16), all f32
  return __builtin_amdgcn_wmma_f32_16x16x4_f32(false, a, false, b, (short)0, c,
                                               false, false);
}

__global__ __launch_bounds__(128) void gru_fused_kernel(
    const float* __restrict__ path,   // (T, BN, 2)
    const float* __restrict__ f_vel,  // (T, BN, 16)
    const float* __restrict__ f_img,  // (B, 32, 80, 80)
    const float* __restrict__ W_ih,   // (144, 52)
    const float* __restrict__ W_hh,   // (144, 48)
    const float* __restrict__ b_ih,   // (144,)
    const float* __restrict__ b_hh,   // (144,)
    float* __restrict__ out)          // outs (T,BN,H) then hT (BN,H)
{
  __shared__ float sWih[G3][20];        // W_ih columns 0..19 (loc/rel/vel)
  __shared__ float sWhh[G3][H];
  __shared__ float sH[4][16][H];        // per-wave h tile
  __shared__ float sImg[IMG_C];
  __shared__ float sPart[IMG_C][4];
  __shared__ float sGi0[G3];            // img@W_ih[:,20:]^T + b_ih
  __shared__ float sSum[T_STEPS][2];    // per-step group loc sums

  const int tid    = threadIdx.x;
  const int wave   = tid >> 5;
  const int lane   = tid & 31;
  const int b      = blockIdx.x;        // block <-> batch element
  const int rowBlk = b * N_GRP;

  // ---- weights -> LDS ----
  for (int i = tid; i < G3 * 20; i += 128)
    sWih[i / 20][i % 20] = W_ih[(i / 20) * DIN + (i % 20)];
  for (int i = tid; i < G3 * H; i += 128)
    sWhh[i / H][i % H] = W_hh[i];

  // ---- image channel means for this b (float4 streamed) ----
  {
    const int c = tid >> 2, p = tid & 3;
    const v4f* fp = (const v4f*)(f_img + ((long)b * IMG_C + c) * IMG_HW);
    float s = 0.f;
    for (int k = p; k < IMG_HW / 4; k += 4) {
      v4f v = fp[k];
      s += v.x + v.y + v.z + v.w;
    }
    sPart[c][p] = s;
  }

  // ---- per-step neighbor-group loc sums (one-time) ----
  for (int i = tid; i < T_STEPS * 2; i += 128) {
    const int t = i >> 1, comp = i & 1;
    float s = 0.f;
    for (int nn = 0; nn < N_GRP; ++nn)
      s += path[((long)t * BN + rowBlk + nn) * 2 + comp];
    sSum[t][comp] = s;
  }
  __syncthreads();

  if (tid < IMG_C)
    sImg[tid] = (sPart[tid][0] + sPart[tid][1] + sPart[tid][2] + sPart[tid][3]) *
                (1.f / (float)IMG_HW);
  __syncthreads();

  // ---- gi0 = img_feat @ W_ih[:,20:52]^T + b_ih (time-invariant) ----
  for (int j = tid; j < G3; j += 128) {
    float g = b_ih[j];
    for (int c = 0; c < IMG_C; ++c) g += sImg[c] * W_ih[j * DIN + 20 + c];
    sGi0[j] = g;
  }
  // zero this wave's h tile
  for (int i = lane; i < 16 * H; i += 32) sH[wave][i / H][i % H] = 0.f;
  __syncthreads();

  // ---- WMMA lane mapping (ISA 16x16x4 f32 layouts) ----
  const int m    = lane & 15;   // A: row; B: column; C: N-column
  const int hi   = lane >> 4;   // lanes 16..31 hold K+2 (A/B), M+8 (C)
  const int koff = hi * 2;
  const int row0 = rowBlk + wave * 16;
  const int rowm = row0 + m;

  // accumulator init values (broadcast per N-column)
  float baseRZ[6], baseIN[3], baseHN[3];
#pragma unroll
  for (int ct = 0; ct < 6; ++ct) baseRZ[ct] = sGi0[ct * 16 + m] + b_hh[ct * 16 + m];
#pragma unroll
  for (int j = 0; j < 3; ++j) {
    baseIN[j] = sGi0[96 + j * 16 + m];
    baseHN[j] = b_hh[96 + j * 16 + m];
  }

  float hreg[8][3];  // this lane's h values: rows v+hi*8, cols j*16+m
#pragma unroll
  for (int v = 0; v < 8; ++v)
#pragma unroll
    for (int j = 0; j < 3; ++j) hreg[v][j] = 0.f;

  for (int t = 0; t < T_STEPS; ++t) {
    // ---- A fragments, x-part (K = 0..19: loc,rel,vel) ----
    v2f ax[5];
    const v2f loc = *(const v2f*)(path + ((long)t * BN + rowm) * 2);
    if (hi == 0) {
      ax[0] = loc;  // k = 0,1 : loc
    } else {        // k = 2,3 : rel = (sum - loc)/63 - loc
      ax[0].x = (sSum[t][0] - loc.x) * (1.f / 63.f) - loc.x;
      ax[0].y = (sSum[t][1] - loc.y) * (1.f / 63.f) - loc.y;
    }
#pragma unroll
    for (int c = 1; c < 5; ++c)
      ax[c] = *(const v2f*)(f_vel + ((long)t * BN + rowm) * 16 + (c - 1) * 4 + koff);

    // ---- A fragments, h-part (K = 0..47) from LDS ----
    v2f ah[12];
#pragma unroll
    for (int c = 0; c < 12; ++c)
      ah[c] = *(const v2f*)(&sH[wave][m][c * 4 + koff]);

    // ---- accumulators: [0..5]=r,z (gi+gh fused), [6..8]=i_n, [9..11]=h_n ----
    v8f acc[12];
#pragma unroll
    for (int ct = 0; ct < 6; ++ct)
#pragma unroll
      for (int e = 0; e < 8; ++e) acc[ct][e] = baseRZ[ct];
#pragma unroll
    for (int j = 0; j < 3; ++j)
#pragma unroll
      for (int e = 0; e < 8; ++e) {
        acc[6 + j][e] = baseIN[j];
        acc[9 + j][e] = baseHN[j];
      }

    // ---- x-part GEMM: 5 K-chunks into tiles 0..8 (r,z fused + i_n) ----
#pragma unroll
    for (int ct = 0; ct < 9; ++ct) {
      const int n = ct * 16 + m;
#pragma unroll
      for (int c = 0; c < 5; ++c) {
        const v2f bf = *(const v2f*)(&sWih[n][c * 4 + koff]);
        acc[ct] = wmma4(ax[c], bf, acc[ct]);
      }
    }
    // ---- h-part GEMM: 12 K-chunks into tiles 0..5 (r,z) and 9..11 (h_n) ----
#pragma unroll
    for (int ct = 0; ct < 9; ++ct) {
      const int n  = ct * 16 + m;
      const int ai = (ct < 6) ? ct : ct + 3;
#pragma unroll
      for (int c = 0; c < 12; ++c) {
        const v2f bf = *(const v2f*)(&sWhh[n][c * 4 + koff]);
        acc[ai] = wmma4(ah[c], bf, acc[ai]);
      }
    }

    // ---- gates + state update + output store ----
#pragma unroll
    for (int v = 0; v < 8; ++v) {
      const int mrow = v + hi * 8;
#pragma unroll
      for (int j = 0; j < 3; ++j) {
        const float r    = 1.f / (1.f + __expf(-acc[j][v]));
        const float z    = 1.f / (1.f + __expf(-acc[3 + j][v]));
        const float cand = tanhf(acc[6 + j][v] + r * acc[9 + j][v]);
        const float hn   = (1.f - z) * cand + z * hreg[v][j];
        hreg[v][j]       = hn;
        const int col    = j * 16 + m;
        sH[wave][mrow][col] = hn;
        out[((long)t * BN + row0 + mrow) * H + col] = hn;
      }
    }
    __syncthreads();  // order LDS h update vs next step's A-frag loads
  }

  // ---- hT (second output, concatenated after outs) ----
#pragma unroll
  for (int v = 0; v < 8; ++v) {
    const int mrow = v + hi * 8;
#pragma unroll
    for (int j = 0; j < 3; ++j)
      out[(long)T_STEPS * BN * H + (long)(row0 + mrow) * H + j * 16 + m] =
          hreg[v][j];
  }
}

extern "C" void kernel_launch(void* const* d_in, const int* in_sizes, int n_in,
                              void* d_out, int out_size, void* d_ws, size_t ws_size,
                              hipStream_t stream) {
  const float* path  = (const float*)d_in[0];
  const float* f_vel = (const float*)d_in[1];
  const float* f_img = (const float*)d_in[2];
  const float* W_ih  = (const float*)d_in[3];
  const float* W_hh  = (const float*)d_in[4];
  const float* b_ih  = (const float*)d_in[5];
  const float* b_hh  = (const float*)d_in[6];
  float* out = (float*)d_out;
  (void)in_sizes; (void)n_in; (void)out_size; (void)d_ws; (void)ws_size;

  hipLaunchKernelGGL(gru_fused_kernel, dim3(B_BATCH), dim3(128), 0, stream,
                     path, f_vel, f_img, W_ih, W_hh, b_ih, b_hh, out);
}